// luong_gate_attention_74474732913162
// MI455X (gfx1250) — compile-verified
//
#include <hip/hip_runtime.h>

#define H 1024
#define B 64
#define S 2048
#define NSPLIT 8              // S-dimension split for the context stream
#define SCHUNK (S / NSPLIT)   // 256 rows per workgroup

typedef __attribute__((ext_vector_type(2))) float v2f;
typedef __attribute__((ext_vector_type(8))) float v8f;

__device__ __forceinline__ float selu_f(float x) {
    const float alpha = 1.6732632423543772f;
    const float scale = 1.0507009873554805f;
    return scale * (x > 0.f ? x : alpha * (__expf(x) - 1.f));
}

// D[m,n] = act( sum_k X(m,k) * W[n,k] + bias[n] )
// X(m,k) = k < K1 ? A[m,k] : A2[m,k-K1]   (handles the concat([h, x]) inputs)
// M = B = 64 rows, N = H = 1024 cols. One wave per 16x16 tile using
// V_WMMA_F32_16X16X4_F32 (A frag: 16x4 in 2 VGPRs, B frag: 4x16 in 2 VGPRs,
// C/D: 16x16 f32 in 8 VGPRs, per ISA 7.12.2 layouts).
template <int ACT>
__global__ void gemm_wmma(const float* __restrict__ A,
                          const float* __restrict__ A2,
                          int K1, int K,
                          const float* __restrict__ W,
                          const float* __restrict__ bias,
                          float* __restrict__ D) {
    const int lane  = threadIdx.x & 31;
    const int wid   = (blockIdx.x * blockDim.x + threadIdx.x) >> 5;
    const int NT    = H / 16;               // 64 column tiles
    const int mtile = wid / NT;             // 0..3
    const int ntile = wid % NT;             // 0..63
    const int half  = lane >> 4;            // 0: K pair {0,1}, 1: K pair {2,3}
    const int mA    = mtile * 16 + (lane & 15);
    const int nB    = ntile * 16 + (lane & 15);
    const int K2    = K - K1;

    v8f acc = {0.f, 0.f, 0.f, 0.f, 0.f, 0.f, 0.f, 0.f};
    for (int k0 = 0; k0 < K; k0 += 4) {
        const int kk = k0 + 2 * half;
        // K1, K are multiples of 4, so (kk, kk+1) never straddle the concat seam.
        const float* xp = (kk < K1) ? (A  + (size_t)mA * K1 + kk)
                                    : (A2 + (size_t)mA * K2 + (kk - K1));
        float2 af = *(const float2*)xp;
        float2 bf = *(const float2*)(W + (size_t)nB * K + kk);
        v2f a;  a[0]  = af.x; a[1]  = af.y;
        v2f bb; bb[0] = bf.x; bb[1] = bf.y;
        acc = __builtin_amdgcn_wmma_f32_16x16x4_f32(false, a, false, bb,
                                                    (short)0, acc, false, false);
    }
    const float bn = bias[nB];
#pragma unroll
    for (int v = 0; v < 8; ++v) {
        const int m = mtile * 16 + half * 8 + v;  // C/D layout: VGPR v -> M = v + 8*half
        float r = acc[v] + bn;
        if (ACT == 1) r = selu_f(r);
        D[(size_t)m * H + nB] = r;
    }
}

// tau[b] = exp(tanh(dot(feed[b,:], W_f2[0,:]) + b_f2))  — one wave per batch row
__global__ void tau_kernel(const float* __restrict__ feed,
                           const float* __restrict__ Wf2,
                           const float* __restrict__ bf2,
                           float* __restrict__ tau) {
    const int lane = threadIdx.x & 31;
    const int b    = (blockIdx.x * blockDim.x + threadIdx.x) >> 5;  // 0..15
    float p = 0.f;
#pragma unroll
    for (int j = 0; j < 8; ++j) {
        float4 a = *(const float4*)(feed + (size_t)b * H + j * 128 + lane * 4);
        float4 w = *(const float4*)(Wf2 + j * 128 + lane * 4);
        p += a.x * w.x + a.y * w.y + a.z * w.z + a.w * w.w;
    }
#pragma unroll
    for (int off = 16; off > 0; off >>= 1) p += __shfl_xor(p, off, 32);
    if (lane == 0) tau[b] = __expf(tanhf(p + bf2[0]));
}

// Fused single-pass attention, partial over an S-chunk.
// Grid: B * NSPLIT workgroups of 8 waves; block (b, split) streams rows
// [split*SCHUNK, (split+1)*SCHUNK) of context[b] exactly once from HBM:
//   raw score -> scores[]; per-wave online softmax with the row kept in VGPRs;
//   waves merged via LDS f32 atomics into an unnormalized partial
//   (M_part, L_part, acc_part[H]) written to workspace.
__global__ void attn_partial_kernel(const float* __restrict__ context,
                                    const float* __restrict__ gamma,
                                    const float* __restrict__ tau,
                                    float* __restrict__ scores,
                                    float* __restrict__ part_ml,
                                    float* __restrict__ part_acc) {
    __shared__ float accbuf[H];
    __shared__ float wm[8], wl[8];
    const int b     = blockIdx.x / NSPLIT;
    const int split = blockIdx.x % NSPLIT;
    const int lane  = threadIdx.x & 31;
    const int warp  = threadIdx.x >> 5;
    const int s0    = split * SCHUNK;
    const int send  = s0 + SCHUNK;

    for (int i = threadIdx.x; i < H; i += blockDim.x) accbuf[i] = 0.f;

    float4 g[8];
#pragma unroll
    for (int j = 0; j < 8; ++j)
        g[j] = *(const float4*)(gamma + (size_t)b * H + j * 128 + lane * 4);
    const float taub = tau[b];

    float  m = -__builtin_inff();
    float  l = 0.f;
    float4 acc[8];
#pragma unroll
    for (int j = 0; j < 8; ++j) acc[j] = make_float4(0.f, 0.f, 0.f, 0.f);

    for (int s = s0 + warp; s < send; s += 8) {
        const float* row = context + ((size_t)b * S + s) * H;
        if (s + 8 < send)  // prefetch this wave's next row (one cacheline per lane)
            __builtin_prefetch(row + (size_t)8 * H + lane * 32, 0, 0);
        float4 c[8];
        float  p = 0.f;
#pragma unroll
        for (int j = 0; j < 8; ++j) {
            c[j] = *(const float4*)(row + j * 128 + lane * 4);
            p += c[j].x * g[j].x + c[j].y * g[j].y + c[j].z * g[j].z + c[j].w * g[j].w;
        }
#pragma unroll
        for (int off = 16; off > 0; off >>= 1) p += __shfl_xor(p, off, 32);
        const float sc = p / taub;
        if (lane == 0) scores[(size_t)b * S + s] = sc;

        const float mn = fmaxf(m, sc);
        const float r  = __expf(m - mn);    // first iter: exp(-inf) = 0
        const float w  = __expf(sc - mn);
        l = l * r + w;
#pragma unroll
        for (int j = 0; j < 8; ++j) {
            acc[j].x = acc[j].x * r + w * c[j].x;
            acc[j].y = acc[j].y * r + w * c[j].y;
            acc[j].z = acc[j].z * r + w * c[j].z;
            acc[j].w = acc[j].w * r + w * c[j].w;
        }
        m = mn;
    }

    if (lane == 0) { wm[warp] = m; wl[warp] = l; }
    __syncthreads();

    float M = -__builtin_inff();
#pragma unroll
    for (int i = 0; i < 8; ++i) M = fmaxf(M, wm[i]);
    float L = 0.f;
#pragma unroll
    for (int i = 0; i < 8; ++i) L += wl[i] * __expf(wm[i] - M);

    const float sw = __expf(m - M);  // rescale this wave's partial to block max
#pragma unroll
    for (int j = 0; j < 8; ++j) {
        const int hbase = j * 128 + lane * 4;
        atomicAdd(&accbuf[hbase + 0], acc[j].x * sw);
        atomicAdd(&accbuf[hbase + 1], acc[j].y * sw);
        atomicAdd(&accbuf[hbase + 2], acc[j].z * sw);
        atomicAdd(&accbuf[hbase + 3], acc[j].w * sw);
    }
    __syncthreads();

    float* pacc = part_acc + (size_t)blockIdx.x * H;
    for (int i = threadIdx.x; i < H; i += blockDim.x) pacc[i] = accbuf[i];
    if (threadIdx.x == 0) {
        part_ml[blockIdx.x * 2]     = M;
        part_ml[blockIdx.x * 2 + 1] = L;
    }
}

// Cross-block softmax combine per batch row:
//   M = max_i M_i;  L = sum_i L_i * e^(M_i - M);
//   c_new[h] = sum_i acc_i[h] * e^(M_i - M) / L
__global__ void attn_merge_kernel(const float* __restrict__ part_ml,
                                  const float* __restrict__ part_acc,
                                  float* __restrict__ ML,
                                  float* __restrict__ c_new) {
    const int b = blockIdx.x;
    float pm[NSPLIT], pl[NSPLIT];
#pragma unroll
    for (int i = 0; i < NSPLIT; ++i) {
        pm[i] = part_ml[(b * NSPLIT + i) * 2];
        pl[i] = part_ml[(b * NSPLIT + i) * 2 + 1];
    }
    float M = -__builtin_inff();
#pragma unroll
    for (int i = 0; i < NSPLIT; ++i) M = fmaxf(M, pm[i]);
    float L = 0.f;
    float w[NSPLIT];
#pragma unroll
    for (int i = 0; i < NSPLIT; ++i) {
        w[i] = __expf(pm[i] - M);
        L += pl[i] * w[i];
    }
    const float invL = 1.f / L;
    for (int hidx = threadIdx.x; hidx < H; hidx += blockDim.x) {
        float v = 0.f;
#pragma unroll
        for (int i = 0; i < NSPLIT; ++i)
            v += part_acc[(size_t)(b * NSPLIT + i) * H + hidx] * w[i];
        c_new[(size_t)b * H + hidx] = v * invL;
    }
    if (threadIdx.x == 0) { ML[b * 2] = M; ML[b * 2 + 1] = L; }
}

// weights[b,s] = exp(score[b,s] - M_b) / L_b  (no re-read of context)
__global__ void weights_kernel(const float* __restrict__ scores,
                               const float* __restrict__ ML,
                               float* __restrict__ weights) {
    const int idx = blockIdx.x * blockDim.x + threadIdx.x;
    const int b   = idx / S;
    weights[idx] = __expf(scores[idx] - ML[b * 2]) * (1.f / ML[b * 2 + 1]);
}

extern "C" void kernel_launch(void* const* d_in, const int* in_sizes, int n_in,
                              void* d_out, int out_size, void* d_ws, size_t ws_size,
                              hipStream_t stream) {
    const float* h    = (const float*)d_in[0];
    const float* c_t  = (const float*)d_in[1];
    const float* ctx  = (const float*)d_in[2];
    const float* W_in = (const float*)d_in[3];
    const float* b_in = (const float*)d_in[4];
    const float* W_f1 = (const float*)d_in[5];
    const float* b_f1 = (const float*)d_in[6];
    const float* W_f2 = (const float*)d_in[7];
    const float* b_f2 = (const float*)d_in[8];
    const float* W_o1 = (const float*)d_in[9];
    const float* b_o1 = (const float*)d_in[10];
    const float* W_o2 = (const float*)d_in[11];
    const float* b_o2 = (const float*)d_in[12];

    // d_out: out [B,H] | weights [B,S] | c_new [B,H]   (reference return order)
    float* out     = (float*)d_out;
    float* weights = out + B * H;
    float* c_new   = weights + B * S;

    float* ws       = (float*)d_ws;
    float* feed     = ws;  ws += B * H;              // selu(hc @ W_f1^T + b_f1)
    float* gamma    = ws;  ws += B * H;              // h @ W_in^T + b_in
    float* t1       = ws;  ws += B * H;              // selu(ho @ W_o1^T + b_o1)
    float* scores   = ws;  ws += B * S;              // raw (pre-softmax) scores
    float* tau      = ws;  ws += B;
    float* ML       = ws;  ws += 2 * B;              // per-batch softmax max / denom
    float* part_ml  = ws;  ws += 2 * B * NSPLIT;     // per-(b,split) partial M, L
    float* part_acc = ws;  ws += (size_t)B * NSPLIT * H;  // partial weighted sums

    dim3 blk(256);  // 8 wave32s

    // 1) feed = selu(concat(h, c_t) @ W_f1^T + b_f1)        [WMMA f32]
    gemm_wmma<1><<<32, blk, 0, stream>>>(h, c_t, H, 2 * H, W_f1, b_f1, feed);
    // 2) tau = exp(tanh(feed @ W_f2^T + b_f2))
    tau_kernel<<<2, blk, 0, stream>>>(feed, W_f2, b_f2, tau);
    // 3) gamma = h @ W_in^T + b_in                           [WMMA f32]
    gemm_wmma<0><<<32, blk, 0, stream>>>(h, h, H, H, W_in, b_in, gamma);
    // 4) single-pass context stream, split 8x for memory-level parallelism
    attn_partial_kernel<<<B * NSPLIT, blk, 0, stream>>>(ctx, gamma, tau, scores,
                                                        part_ml, part_acc);
    // 5) cross-block softmax combine -> c_new, (M, L)
    attn_merge_kernel<<<B, blk, 0, stream>>>(part_ml, part_acc, ML, c_new);
    // 6) weights from stored raw scores
    weights_kernel<<<(B * S) / 256, blk, 0, stream>>>(scores, ML, weights);
    // 7) t1 = selu(concat(h, c_new) @ W_o1^T + b_o1)         [WMMA f32]
    gemm_wmma<1><<<32, blk, 0, stream>>>(h, c_new, H, 2 * H, W_o1, b_o1, t1);
    // 8) out = selu(t1 @ W_o2^T + b_o2)                      [WMMA f32]
    gemm_wmma<1><<<32, blk, 0, stream>>>(t1, t1, H, H, W_o2, b_o2, out);
}